// NodeNetGNN_25855703122293
// MI455X (gfx1250) — compile-verified
//
#include <hip/hip_runtime.h>
#include <hip/hip_bf16.h>

typedef __attribute__((ext_vector_type(2))) float v2f;
typedef __attribute__((ext_vector_type(8))) float v8f;

#define NN 100000   // N_NODE
#define NT 100000   // N_NET
#define NE 400000   // N_EDGE
// H_NODE=32, H_NET=16, H_PIN=8, O_NODE=16, O_NET=32

// ---------------------------------------------------------------- zero init
__global__ void gnn_zero(float* out, long nOut, float* ws, long nWs) {
    long i = (long)blockIdx.x * blockDim.x + threadIdx.x;
    long stride = (long)gridDim.x * blockDim.x;
    for (long j = i; j < nOut; j += stride) out[j] = 0.0f;
    for (long j = i; j < nWs;  j += stride) ws[j]  = 0.0f;
}

// ---------------------------------------------------------------- degrees
__global__ void gnn_degrees(const int* __restrict__ src, const int* __restrict__ dst,
                            float* degNode, float* degNet) {
    int e = blockIdx.x * blockDim.x + threadIdx.x;
    if (e >= NE) return;
    atomicAdd(&degNode[src[e]], 1.0f);
    atomicAdd(&degNet[dst[e]], 1.0f);
}

// ------------------------------------------- node -> net normalized scatter
// agg[dst][f] += node_feat[src][f] * rsqrt(max(out_deg[src],1))
__global__ void gnn_scatter_net(const float* __restrict__ nodeFeat,
                                const int* __restrict__ src, const int* __restrict__ dst,
                                const float* __restrict__ degNode,
                                float* agg /* = net_out region of d_out */) {
    long idx = (long)blockIdx.x * blockDim.x + threadIdx.x;
    if (idx >= (long)NE * 32) return;
    int e = (int)(idx >> 5);
    int f = (int)(idx & 31);
    int s = src[e];
    float rs = __frsqrt_rn(fmaxf(degNode[s], 1.0f));
    float v = nodeFeat[(long)s * 32 + f] * rs;
    atomicAdd(&agg[(long)dst[e] * 32 + f], v);
}

// ------------------------------------------------ net GEMM (in place, WMMA)
// net_out = (agg * rsqrt(in_deg)) @ gc_weight + gc_bias    (N_NET,32)x(32,32)
__global__ void gnn_net_gemm(float* netOut, const float* __restrict__ degNet,
                             const float* __restrict__ gcW, const float* __restrict__ gcB) {
    int wave = threadIdx.x >> 5;
    int lane = threadIdx.x & 31;
    int tile = blockIdx.x * (blockDim.x >> 5) + wave;
    if (tile >= NT / 16) return;            // wave-uniform; EXEC stays all-ones
    int hi   = lane >> 4;                   // half-wave selects K pair
    int lrow = lane & 15;
    int row  = tile * 16 + lrow;
    float rs = __frsqrt_rn(fmaxf(degNet[row], 1.0f));
    const float* arow = netOut + (long)row * 32;

    v8f c0 = {}; v8f c1 = {};
#pragma unroll
    for (int kk = 0; kk < 8; ++kk) {
        int k0 = 4 * kk + 2 * hi;
        v2f a, b0, b1;
        a.x = arow[k0] * rs;
        a.y = arow[k0 + 1] * rs;
        b0.x = gcW[k0 * 32 + lrow];        b0.y = gcW[(k0 + 1) * 32 + lrow];
        b1.x = gcW[k0 * 32 + 16 + lrow];   b1.y = gcW[(k0 + 1) * 32 + 16 + lrow];
        c0 = __builtin_amdgcn_wmma_f32_16x16x4_f32(false, a, false, b0, (short)0, c0, false, false);
        c1 = __builtin_amdgcn_wmma_f32_16x16x4_f32(false, a, false, b1, (short)0, c1, false, false);
    }
    float bias0 = gcB[lrow], bias1 = gcB[16 + lrow];
#pragma unroll
    for (int r = 0; r < 8; ++r) {
        int orow = tile * 16 + r + 8 * hi;
        netOut[(long)orow * 32 + lrow]      = c0[r] + bias0;
        netOut[(long)orow * 32 + 16 + lrow] = c1[r] + bias1;
    }
}

// ----------------------------- NNConv edge messages as (E,144)x(144,16) GEMM
// Z[e, 8i+p] = net_feat[dst[e]][i] * pin_feat[e][p]   (k in [0,128))
// Z[e, 128+i] = net_feat[dst[e]][i]                   (bias fold, k in [128,144))
// Wext[8i+p][o] = lin_W[p][16i+o] ;  Wext[128+i][o] = lin_b[16i+o]
// scatter: nodeAcc[src[e]][o] += m[e][o]
__global__ void gnn_edge_wmma(const float* __restrict__ netFeat,
                              const float* __restrict__ pinFeat,
                              const int* __restrict__ pinsSrc,
                              const int* __restrict__ pinsDst,
                              const float* __restrict__ linW,
                              const float* __restrict__ linB,
                              float* nodeAcc /* = node_out region of d_out */) {
    __shared__ float sW[2048];       // lin_W (8 x 256)
    __shared__ float sB[256];        // lin_b (16 x 16)
    __shared__ float sSrc[8][256];   // per-wave: 16 edges x 16 net feats
    __shared__ float sPin[8][128];   // per-wave: 16 edges x 8 pin feats
    __shared__ int   sIdx[8][16];    // per-wave: 16 dest node ids

    int tid = threadIdx.x;
    for (int i = tid; i < 2048; i += 256) sW[i] = linW[i];
    if (tid < 256) sB[tid] = linB[tid];

    int wave = tid >> 5;
    int lane = tid & 31;
    int tile = blockIdx.x * 8 + wave;      // 25000 tiles = 3125 blocks * 8 waves
    int er = lane & 15;                    // edge-in-tile / output column
    int hi = lane >> 4;
    long e = (long)tile * 16 + er;

    if (hi == 0) {
        const float4* nf = (const float4*)(netFeat + (long)pinsDst[e] * 16);
        float4* d = (float4*)&sSrc[wave][er * 16];
        d[0] = nf[0]; d[1] = nf[1]; d[2] = nf[2]; d[3] = nf[3];
        sIdx[wave][er] = pinsSrc[e];
    } else {
        const float4* pf = (const float4*)(pinFeat + e * 8);
        float4* d = (float4*)&sPin[wave][er * 8];
        d[0] = pf[0]; d[1] = pf[1];
    }
    __syncthreads();

    const float* mySrc = &sSrc[wave][er * 16];
    const float* myPin = &sPin[wave][er * 8];

    v8f c = {};
#pragma unroll
    for (int kk = 0; kk < 32; ++kk) {      // k < 128 : outer-product part
        int k0 = 4 * kk + 2 * hi;
        v2f a, b;
        a.x = mySrc[k0 >> 3]       * myPin[k0 & 7];
        a.y = mySrc[(k0 + 1) >> 3] * myPin[(k0 + 1) & 7];
        b.x = sW[(k0 & 7) * 256 + (k0 >> 3) * 16 + er];
        b.y = sW[((k0 + 1) & 7) * 256 + ((k0 + 1) >> 3) * 16 + er];
        c = __builtin_amdgcn_wmma_f32_16x16x4_f32(false, a, false, b, (short)0, c, false, false);
    }
#pragma unroll
    for (int kk = 32; kk < 36; ++kk) {     // k in [128,144) : bias fold
        int k0 = 4 * kk + 2 * hi - 128;
        v2f a, b;
        a.x = mySrc[k0];     a.y = mySrc[k0 + 1];
        b.x = sB[k0 * 16 + er];
        b.y = sB[(k0 + 1) * 16 + er];
        c = __builtin_amdgcn_wmma_f32_16x16x4_f32(false, a, false, b, (short)0, c, false, false);
    }
#pragma unroll
    for (int r = 0; r < 8; ++r) {
        int nid = sIdx[wave][r + 8 * hi];
        atomicAdd(&nodeAcc[(long)nid * 16 + er], c[r]);
    }
}

// ---------------------------------------------------------------- finalize
__global__ void gnn_node_final(float* nodeOut, const float* __restrict__ degNode,
                               const float* __restrict__ nnBias) {
    long i = (long)blockIdx.x * blockDim.x + threadIdx.x;
    if (i >= (long)NN * 16) return;
    int n = (int)(i >> 4);
    int o = (int)(i & 15);
    nodeOut[i] = nodeOut[i] / fmaxf(degNode[n], 1.0f) + nnBias[o];
}

extern "C" void kernel_launch(void* const* d_in, const int* in_sizes, int n_in,
                              void* d_out, int out_size, void* d_ws, size_t ws_size,
                              hipStream_t stream) {
    const float* node_feat = (const float*)d_in[0];
    const float* net_feat  = (const float*)d_in[1];
    const float* pin_feat  = (const float*)d_in[2];
    const int*   pins_src  = (const int*)d_in[3];
    const int*   pins_dst  = (const int*)d_in[4];
    const float* gc_weight = (const float*)d_in[5];
    const float* gc_bias   = (const float*)d_in[6];
    const float* lin_W     = (const float*)d_in[7];
    const float* lin_b     = (const float*)d_in[8];
    const float* nn_bias   = (const float*)d_in[9];

    float* out      = (float*)d_out;
    float* node_out = out;                    // (N_NODE, 16)  first in tuple order
    float* net_out  = out + (long)NN * 16;    // (N_NET, 32)

    float* degNode = (float*)d_ws;            // N_NODE floats
    float* degNet  = degNode + NN;            // N_NET  floats

    // 1. zero accumulators (d_out doubles as the accumulation buffers)
    gnn_zero<<<2048, 256, 0, stream>>>(out, (long)NN * 16 + (long)NT * 32,
                                       degNode, (long)NN + NT);
    // 2. degrees
    gnn_degrees<<<(NE + 255) / 256, 256, 0, stream>>>(pins_src, pins_dst, degNode, degNet);
    // 3. node -> net normalized scatter into net_out (acts as 'agg')
    {
        long total = (long)NE * 32;
        gnn_scatter_net<<<(int)((total + 255) / 256), 256, 0, stream>>>(
            node_feat, pins_src, pins_dst, degNode, net_out);
    }
    // 4. net GEMM in place (WMMA f32 16x16x4), 8 waves/block, 6250 tiles
    gnn_net_gemm<<<(NT / 16 + 7) / 8, 256, 0, stream>>>(net_out, degNet, gc_weight, gc_bias);
    // 5. edge NNConv as WMMA GEMM + atomic scatter, 25000 tiles / 8 waves = 3125 blocks
    gnn_edge_wmma<<<NE / 16 / 8, 256, 0, stream>>>(net_feat, pin_feat, pins_src, pins_dst,
                                                   lin_W, lin_b, node_out);
    // 6. finalize node output: mean + bias
    {
        long total = (long)NN * 16;
        gnn_node_final<<<(int)((total + 255) / 256), 256, 0, stream>>>(node_out, degNode, nn_bias);
    }
}